// NormalizedDynamicsOptimized_57561151701123
// MI455X (gfx1250) — compile-verified
//
#include <hip/hip_runtime.h>
#include <math.h>

typedef __attribute__((ext_vector_type(2))) float v2f;
typedef __attribute__((ext_vector_type(8))) float v8f;

#define NROW 8192
#define NDIM 16
#define NTILE (NROW / 16)     // 512
#define NSPLIT 8              // j-dimension split per row tile
#define JT_PER_SPLIT (NTILE / NSPLIT)  // 64
#define EPSV 1e-8f
#define NOISEF 0.01f
#define BIGF 3.4028235e38f
#define LOG2E 1.4426950408889634f

// workspace layout (float offsets)
#define OFF_XC     0          // 131072
#define OFF_H      131072     // 131072
#define OFF_SQ     262144     // 8192
#define OFF_SIGMA  270336     // 8192
#define OFF_S1P    278528     // 8*8192
#define OFF_S2P    344064     // 8*8192
#define OFF_DRIFTP 409600     // 8*131072
#define OFF_RWP    1458176    // 8*8192
#define OFF_MEANX  1523712    // 16
#define OFF_STDX   1523728    // 16
#define OFF_SCALE  1523744    // 16
#define OFF_K      1523760    // 1 int

__device__ __forceinline__ float fsqrt_fast(float x) { return __builtin_amdgcn_sqrtf(x); }
__device__ __forceinline__ float fexp2_fast(float x) { return __builtin_amdgcn_exp2f(x); }
__device__ __forceinline__ float frcp_fast(float x)  { return __builtin_amdgcn_rcpf(x); }

__device__ __forceinline__ v8f wmma4(v2f a, v2f b, v8f c) {
    // D = A(16x4) * B(4x16) + C(16x16), f32
    return __builtin_amdgcn_wmma_f32_16x16x4_f32(false, a, false, b,
                                                 (short)0, c, false, false);
}

// A/B operand registers for a 16x16 f32 tile of xc (rows base..base+15, dims 0..15)
// as four chained K=4 WMMAs. lane l holds M(or N)=l&15 ; K = 4q + 2*(l>>4) + {0,1}
__device__ __forceinline__ void load_rowop(const float* __restrict__ xc, int base,
                                           int lane, v2f op[4]) {
    const float* row = xc + (base + (lane & 15)) * NDIM;
    int h2 = (lane >> 4) << 1;
#pragma unroll
    for (int q = 0; q < 4; ++q) {
        op[q].x = row[4 * q + h2];
        op[q].y = row[4 * q + h2 + 1];
    }
}

// ---------------- K0: column mean / std of x ----------------
__global__ void col_stats_x(const float* __restrict__ x,
                            float* __restrict__ meanx, float* __restrict__ stdx) {
    __shared__ float sm1[256], sm2[256];
    int t = threadIdx.x, col = t & 15, grp = t >> 4;
    float s1 = 0.f, s2 = 0.f;
    for (int r = grp; r < NROW; r += 16) {
        float v = x[r * NDIM + col];
        s1 += v; s2 += v * v;
    }
    sm1[t] = s1; sm2[t] = s2;
    __syncthreads();
    if (grp == 0) {
        float a1 = 0.f, a2 = 0.f;
        for (int g = 0; g < 16; ++g) { a1 += sm1[g * 16 + col]; a2 += sm2[g * 16 + col]; }
        float mean = a1 / (float)NROW;
        float var = (a2 - a1 * a1 / (float)NROW) / (float)(NROW - 1);
        meanx[col] = mean;
        stdx[col] = sqrtf(fmaxf(var, 0.f));
    }
}

// ---------------- K0b: xc = x - mean, sq = ||xc||^2 ----------------
__global__ void center_rows(const float* __restrict__ x, const float* __restrict__ meanx,
                            float* __restrict__ xc, float* __restrict__ sq) {
    int row = blockIdx.x * blockDim.x + threadIdx.x;
    if (row >= NROW) return;
    float s = 0.f;
#pragma unroll
    for (int c = 0; c < NDIM; ++c) {
        float v = x[row * NDIM + c] - meanx[c];
        xc[row * NDIM + c] = v;
        s += v * v;
    }
    sq[row] = s;
}

// ------- K1: WMMA pass A — per-row sum / sumsq of distances (split partials) -------
__global__ void row_dist_stats(const float* __restrict__ xc, const float* __restrict__ sq,
                               float* __restrict__ s1p, float* __restrict__ s2p) {
    int lane = threadIdx.x & 31;
    int wave = threadIdx.x >> 5;
    int wid  = blockIdx.x * 8 + wave;          // 0..4095
    int tile = wid >> 3;                       // 0..511
    int split = wid & (NSPLIT - 1);            // 0..7
    int rowbase = tile * 16;
    int h = lane >> 4;
    int nsel = lane & 15;

    v2f a[4];
    load_rowop(xc, rowbase, lane, a);
    float sqA[8];
#pragma unroll
    for (int r = 0; r < 8; ++r) sqA[r] = sq[rowbase + r + 8 * h];

    float s1[8] = {}, s2[8] = {};
    int jt0 = split * JT_PER_SPLIT;
    for (int jt = jt0; jt < jt0 + JT_PER_SPLIT; ++jt) {
        int jbase = jt * 16;
        // unconditional prefetch of next j-tile, wrapped (stays inside xc)
        __builtin_prefetch(xc + (((jbase + 16) & (NROW - 1)) * NDIM), 0, 1);
        v2f b[4];
        load_rowop(xc, jbase, lane, b);
        v8f acc = {};
#pragma unroll
        for (int q = 0; q < 4; ++q) acc = wmma4(a[q], b[q], acc);
        float sqB = sq[jbase + nsel];
#pragma unroll
        for (int r = 0; r < 8; ++r) {
            float d2 = sqA[r] + sqB - 2.0f * acc[r];
            float dist = fsqrt_fast(fmaxf(d2, 0.f));
            s1[r] += dist;
            s2[r] += dist * dist;
        }
    }
#pragma unroll
    for (int r = 0; r < 8; ++r) {
        float a1 = s1[r], a2 = s2[r];
#pragma unroll
        for (int m = 1; m <= 8; m <<= 1) {
            a1 += __shfl_xor(a1, m, 32);
            a2 += __shfl_xor(a2, m, 32);
        }
        if (nsel == r) {                       // lanes 0..7 / 16..23 cover 16 rows
            int row = rowbase + r + 8 * h;
            s1p[split * NROW + row] = a1;
            s2p[split * NROW + row] = a2;
        }
    }
}

// ---------------- K2: scalar adaptive k (reduce split partials) ----------------
__global__ void compute_k(const float* __restrict__ s1p, const float* __restrict__ s2p,
                          int* __restrict__ kout) {
    __shared__ float red[256];
    __shared__ float gmaxs;
    int t = threadIdx.x;
    float lmax = 0.f;
    for (int i = t; i < NROW; i += 256) {
        float a1 = 0.f, a2 = 0.f;
#pragma unroll
        for (int s = 0; s < NSPLIT; ++s) { a1 += s1p[s * NROW + i]; a2 += s2p[s * NROW + i]; }
        float var = (a2 - a1 * a1 / (float)NROW) / (float)(NROW - 1);
        lmax = fmaxf(lmax, fsqrt_fast(fmaxf(var, 0.f)));
    }
    red[t] = lmax;
    __syncthreads();
    for (int s = 128; s > 0; s >>= 1) {
        if (t < s) red[t] = fmaxf(red[t], red[t + s]);
        __syncthreads();
    }
    if (t == 0) gmaxs = red[0];
    __syncthreads();
    float inv = 1.f / (gmaxs + EPSV);
    float lsum = 0.f;
    for (int i = t; i < NROW; i += 256) {
        float a1 = 0.f, a2 = 0.f;
#pragma unroll
        for (int s = 0; s < NSPLIT; ++s) { a1 += s1p[s * NROW + i]; a2 += s2p[s * NROW + i]; }
        float var = (a2 - a1 * a1 / (float)NROW) / (float)(NROW - 1);
        float den = fsqrt_fast(fmaxf(var, 0.f));
        float ka = 5.f + 10.f * den * inv;
        ka = fminf(fmaxf(ka, 5.f), 20.f);
        lsum += (float)((int)ka);               // astype(int32) truncation
    }
    __syncthreads();
    red[t] = lsum;
    __syncthreads();
    for (int s = 128; s > 0; s >>= 1) {
        if (t < s) red[t] += red[t + s];
        __syncthreads();
    }
    if (t == 0) *kout = (int)(red[0] / (float)NROW);  // int() truncation
}

// ------- K3: per-row k-th smallest distance (sigma), one wave per row -------
__global__ void select_sigma(const float* __restrict__ xc, const float* __restrict__ sq,
                             const int* __restrict__ kptr, float* __restrict__ sigma) {
    int lane = threadIdx.x & 31;
    int wave = threadIdx.x >> 5;
    int row = blockIdx.x * 8 + wave;
    int k = *kptr;

    const float4* rp = (const float4*)(xc + row * NDIM);
    float4 r0 = rp[0], r1 = rp[1], r2 = rp[2], r3 = rp[3];
    float sqi = sq[row];

    float lst[20];
#pragma unroll
    for (int t = 0; t < 20; ++t) lst[t] = BIGF;

    for (int jb = 0; jb < NROW; jb += 32) {
        int j = jb + lane;
        const float4* jp = (const float4*)(xc + j * NDIM);
        float4 a0 = jp[0], a1 = jp[1], a2 = jp[2], a3 = jp[3];
        float dot = r0.x * a0.x + r0.y * a0.y + r0.z * a0.z + r0.w * a0.w
                  + r1.x * a1.x + r1.y * a1.y + r1.z * a1.z + r1.w * a1.w
                  + r2.x * a2.x + r2.y * a2.y + r2.z * a2.z + r2.w * a2.w
                  + r3.x * a3.x + r3.y * a3.y + r3.z * a3.z + r3.w * a3.w;
        float d2 = sqi + sq[j] - 2.f * dot;
        float dist = fsqrt_fast(fmaxf(d2, 0.f));
        if (dist < lst[19]) {                  // branchless sorted insert (asc)
            float cur = dist;
#pragma unroll
            for (int t = 0; t < 20; ++t) {
                float lo = fminf(cur, lst[t]);
                float hi = fmaxf(cur, lst[t]);
                lst[t] = lo;
                cur = hi;
            }
        }
    }
    // 32-way merge: pop global min k times; k <= 20
    float kth = 0.f;
    for (int t = 0; t < k; ++t) {
        float v = lst[0];
        float m = v;
#pragma unroll
        for (int mk = 1; mk <= 16; mk <<= 1) m = fminf(m, __shfl_xor(m, mk, 32));
        unsigned long long bal = __ballot(v == m);
        int first = __ffsll(bal) - 1;
        if (lane == first) {
#pragma unroll
            for (int u = 0; u < 19; ++u) lst[u] = lst[u + 1];
            lst[19] = BIGF;
        }
        kth = m;
    }
    if (lane == 0) sigma[row] = kth;
}

// ------- K4: WMMA pass B — exp-kernel + drift partials (split, no atomics) -------
__global__ void drift_part(const float* __restrict__ xc, const float* __restrict__ sq,
                           const float* __restrict__ sigma,
                           float* __restrict__ driftp, float* __restrict__ rwp) {
    __shared__ float wsm[8][16 * 18];          // per-wave 16x16 tile, pitch 18
    int lane = threadIdx.x & 31;
    int wave = threadIdx.x >> 5;
    int wid  = blockIdx.x * 8 + wave;          // 0..4095
    int tile = wid >> 3;
    int split = wid & (NSPLIT - 1);
    int rowbase = tile * 16;
    int h = lane >> 4;
    int nsel = lane & 15;
    float* wl = wsm[wave];

    v2f a[4];
    load_rowop(xc, rowbase, lane, a);
    float sqA[8], invden[8];
#pragma unroll
    for (int r = 0; r < 8; ++r) {
        int row = rowbase + r + 8 * h;
        sqA[r] = sq[row];
        float sg = sigma[row];
        invden[r] = LOG2E / (2.0f * sg * sg + EPSV);  // exp(-d*c) = exp2(-d*c*log2e)
    }

    v8f drift = {};
    float accW[8] = {};
    int jt0 = split * JT_PER_SPLIT;
    for (int jt = jt0; jt < jt0 + JT_PER_SPLIT; ++jt) {
        int jbase = jt * 16;
        __builtin_prefetch(xc + (((jbase + 16) & (NROW - 1)) * NDIM), 0, 1);
        v2f b[4];
        load_rowop(xc, jbase, lane, b);
        v8f s = {};
#pragma unroll
        for (int q = 0; q < 4; ++q) s = wmma4(a[q], b[q], s);
        float sqB = sq[jbase + nsel];
#pragma unroll
        for (int r = 0; r < 8; ++r) {
            float d2 = sqA[r] + sqB - 2.f * s[r];
            float dist = fsqrt_fast(fmaxf(d2, 0.f));
            float w = fexp2_fast(-dist * invden[r]);
            accW[r] += w;
            wl[(r + 8 * h) * 18 + nsel] = w;   // C-layout -> row-major LDS
        }
        // reload W in A-layout, load xc j-tile in B-layout (transposed access)
        v2f aw[4], bx[4];
        int h2 = h << 1;
#pragma unroll
        for (int q = 0; q < 4; ++q) {
            int kk = 4 * q + h2;
            aw[q].x = wl[nsel * 18 + kk];
            aw[q].y = wl[nsel * 18 + kk + 1];
            bx[q].x = xc[(jbase + kk) * NDIM + nsel];
            bx[q].y = xc[(jbase + kk + 1) * NDIM + nsel];
        }
#pragma unroll
        for (int q = 0; q < 4; ++q) drift = wmma4(aw[q], bx[q], drift);
    }

#pragma unroll
    for (int r = 0; r < 8; ++r) {
        int row = rowbase + r + 8 * h;
        driftp[split * (NROW * NDIM) + row * NDIM + nsel] = drift[r];
        float rw = accW[r];
#pragma unroll
        for (int m = 1; m <= 8; m <<= 1) rw += __shfl_xor(rw, m, 32);
        if (nsel == r) rwp[split * NROW + row] = rw;
    }
}

// ------- K4b: reduce drift partials, apply h update -------
__global__ void h_update(const float* __restrict__ xc, const float* __restrict__ noise,
                         const float* __restrict__ alpha,
                         const float* __restrict__ driftp, const float* __restrict__ rwp,
                         float* __restrict__ hout) {
    int idx = blockIdx.x * blockDim.x + threadIdx.x;
    if (idx >= NROW * NDIM) return;
    int row = idx >> 4;
    float dsum = 0.f, rw = 0.f;
#pragma unroll
    for (int s = 0; s < NSPLIT; ++s) {
        dsum += driftp[s * (NROW * NDIM) + idx];
        rw   += rwp[s * NROW + row];
    }
    float step = exp2f(-4.0f * alpha[0]);       // D^-alpha = 16^-alpha
    float xce = xc[idx];
    float dr = dsum * frcp_fast(rw + EPSV);
    hout[idx] = xce + step * (dr - xce) + noise[idx] * NOISEF;
}

// ---------------- K5: column std of h -> scale ----------------
__global__ void col_stats_h(const float* __restrict__ hbuf, const float* __restrict__ stdx,
                            float* __restrict__ scale) {
    __shared__ float sm1[256], sm2[256];
    int t = threadIdx.x, col = t & 15, grp = t >> 4;
    float s1 = 0.f, s2 = 0.f;
    for (int r = grp; r < NROW; r += 16) {
        float v = hbuf[r * NDIM + col];
        s1 += v; s2 += v * v;
    }
    sm1[t] = s1; sm2[t] = s2;
    __syncthreads();
    if (grp == 0) {
        float a1 = 0.f, a2 = 0.f;
        for (int g = 0; g < 16; ++g) { a1 += sm1[g * 16 + col]; a2 += sm2[g * 16 + col]; }
        float var = (a2 - a1 * a1 / (float)NROW) / (float)(NROW - 1);
        float hstd = sqrtf(fmaxf(var, 0.f));
        scale[col] = stdx[col] / (hstd + EPSV);
    }
}

// ---------------- K6: out = h * scale + mean ----------------
__global__ void finalize(const float* __restrict__ hbuf, const float* __restrict__ scale,
                         const float* __restrict__ meanx, float* __restrict__ out) {
    int idx = blockIdx.x * blockDim.x + threadIdx.x;
    if (idx >= NROW * NDIM) return;
    int c = idx & 15;
    out[idx] = hbuf[idx] * scale[c] + meanx[c];
}

extern "C" void kernel_launch(void* const* d_in, const int* in_sizes, int n_in,
                              void* d_out, int out_size, void* d_ws, size_t ws_size,
                              hipStream_t stream) {
    const float* x     = (const float*)d_in[0];
    const float* noise = (const float*)d_in[1];
    const float* alpha = (const float*)d_in[2];
    float* out = (float*)d_out;
    float* ws  = (float*)d_ws;

    float* xc     = ws + OFF_XC;
    float* hbuf   = ws + OFF_H;
    float* sq     = ws + OFF_SQ;
    float* sigma  = ws + OFF_SIGMA;
    float* s1p    = ws + OFF_S1P;
    float* s2p    = ws + OFF_S2P;
    float* driftp = ws + OFF_DRIFTP;
    float* rwp    = ws + OFF_RWP;
    float* meanx  = ws + OFF_MEANX;
    float* stdx   = ws + OFF_STDX;
    float* scale  = ws + OFF_SCALE;
    int*   kptr   = (int*)(ws + OFF_K);

    col_stats_x<<<1, 256, 0, stream>>>(x, meanx, stdx);
    center_rows<<<NROW / 256, 256, 0, stream>>>(x, meanx, xc, sq);
    row_dist_stats<<<(NTILE * NSPLIT) / 8, 256, 0, stream>>>(xc, sq, s1p, s2p);
    compute_k<<<1, 256, 0, stream>>>(s1p, s2p, kptr);
    select_sigma<<<NROW / 8, 256, 0, stream>>>(xc, sq, kptr, sigma);
    drift_part<<<(NTILE * NSPLIT) / 8, 256, 0, stream>>>(xc, sq, sigma, driftp, rwp);
    h_update<<<(NROW * NDIM) / 256, 256, 0, stream>>>(xc, noise, alpha, driftp, rwp, hbuf);
    col_stats_h<<<1, 256, 0, stream>>>(hbuf, stdx, scale);
    finalize<<<(NROW * NDIM) / 256, 256, 0, stream>>>(hbuf, scale, meanx, out);
}